// MambaClassifier_70119636075018
// MI455X (gfx1250) — compile-verified
//
#include <hip/hip_runtime.h>
#include <hip/hip_bf16.h>
#include <math.h>

// ---------------- Model dims (match reference) ----------------
#define D_MODELX   768
#define N_LAYERSX  4
#define NUM_CLS    10
#define D_INNERX   1536
#define D_STATEX   16
#define D_CONVX    4
#define DT_RANKX   48
#define BATCHX     8
#define SEQX       2048
#define M_ROWS     (BATCHX * SEQX)     // 16384 rows for all GEMMs
#define EPSF       1e-5f
#define DBC_W      (DT_RANKX + 2 * D_STATEX)   // 80
#define SCAN_CHUNK 16                  // timesteps of B/C staged per barrier

typedef float v2f __attribute__((ext_vector_type(2)));
typedef float v8f __attribute__((ext_vector_type(8)));

enum { EPI_STORE = 0, EPI_SOFTPLUS_BIAS = 1, EPI_RESID = 2 };

// ---------------------------------------------------------------------------
// fp32 WMMA GEMM:  C[M x N] = A[M x K] * W^T   (W is N x K row-major)
// Each wave owns a 16 x (16*NT) strip: one A fragment feeds NT WMMAs per
// k-step (A-operand reuse), and each N-tile keeps dual-K accumulators so
// 2*NT independent V_WMMA_F32_16X16X4_F32 chains stay in flight.
// 8 waves (256 thr) stack along M.
// Fragment layout per ISA 7.12.2:
//   A 16x4 : lanes 0-15 rows M, VGPR{0,1}=K{0,1}; lanes 16-31 K{2,3} -> float2
//   B 4x16 : mirror over rows of W (row n of W supplies column n)
//   C/D    : VGPR r -> M = r + 8*(lane>=16), N = lane&15
// ---------------------------------------------------------------------------
template<int EPI, int NT>
__global__ __launch_bounds__(256)
void gemm_wmma_f32(const float* __restrict__ A, int lda,
                   const float* __restrict__ W, int ldw,
                   float* __restrict__ C, int ldc,
                   int K, const float* __restrict__ bias)
{
    const int lane = threadIdx.x;              // 0..31 (wave32)
    const int wv   = threadIdx.y;              // 0..7
    const int m0   = (blockIdx.x * 8 + wv) * 16;
    const int n0   = blockIdx.y * (16 * NT);
    const int hl   = lane >> 4;                // 0 / 1
    const int lm   = lane & 15;

    const float* arow = A + (size_t)(m0 + lm) * lda + 2 * hl;
    const float* brow[NT];
#pragma unroll
    for (int j = 0; j < NT; ++j)
        brow[j] = W + (size_t)(n0 + 16 * j + lm) * ldw + 2 * hl;

    v8f cA[NT];
    v8f cB[NT];
#pragma unroll
    for (int j = 0; j < NT; ++j) { cA[j] = {}; cB[j] = {}; }

    for (int k0 = 0; k0 < K; k0 += 8) {
        v2f a0 = *(const v2f*)(arow + k0);
        v2f a1 = *(const v2f*)(arow + k0 + 4);
#pragma unroll
        for (int j = 0; j < NT; ++j) {
            v2f b0 = *(const v2f*)(brow[j] + k0);
            v2f b1 = *(const v2f*)(brow[j] + k0 + 4);
            cA[j] = __builtin_amdgcn_wmma_f32_16x16x4_f32(false, a0, false, b0,
                                                          (short)0, cA[j], false, false);
            cB[j] = __builtin_amdgcn_wmma_f32_16x16x4_f32(false, a1, false, b1,
                                                          (short)0, cB[j], false, false);
        }
    }

#pragma unroll
    for (int j = 0; j < NT; ++j) {
        v8f c = cA[j] + cB[j];
        const int n  = n0 + 16 * j + lm;
        float   bn   = (EPI == EPI_SOFTPLUS_BIAS) ? bias[n] : 0.0f;
#pragma unroll
        for (int r = 0; r < 8; ++r) {
            int    m   = m0 + r + 8 * hl;
            size_t off = (size_t)m * ldc + n;
            float  v   = c[r];
            if (EPI == EPI_STORE) {
                C[off] = v;
            } else if (EPI == EPI_SOFTPLUS_BIAS) {
                v += bn;                       // softplus(v) = log1p(exp(v))
                C[off] = (v > 20.0f) ? v : log1pf(__expf(v));
            } else {                           // residual accumulate
                C[off] += v;
            }
        }
    }
}

// ---------------------------------------------------------------------------
// RMSNorm over rows of 768.  One block (256 thr) per row, 3 elems/thread.
// ---------------------------------------------------------------------------
__global__ __launch_bounds__(256)
void rmsnorm_kernel(const float* __restrict__ X, const float* __restrict__ w,
                    float* __restrict__ Y)
{
    __shared__ float red[8];
    const int    row = blockIdx.x;
    const float* x   = X + (size_t)row * D_MODELX;

    float v[3];
    float s = 0.0f;
#pragma unroll
    for (int i = 0; i < 3; ++i) {
        v[i] = x[threadIdx.x + 256 * i];
        s += v[i] * v[i];
    }
    for (int off = 16; off > 0; off >>= 1) s += __shfl_down(s, off, 32);
    if ((threadIdx.x & 31) == 0) red[threadIdx.x >> 5] = s;
    __syncthreads();
    if (threadIdx.x < 8) {
        float t = red[threadIdx.x];
        for (int off = 4; off > 0; off >>= 1) t += __shfl_down(t, off, 8);
        if (threadIdx.x == 0) red[0] = t;
    }
    __syncthreads();
    const float scale = rsqrtf(red[0] * (1.0f / (float)D_MODELX) + EPSF);
    float* y = Y + (size_t)row * D_MODELX;
#pragma unroll
    for (int i = 0; i < 3; ++i) {
        int d = threadIdx.x + 256 * i;
        y[d] = v[i] * scale * w[d];
    }
}

// ---------------------------------------------------------------------------
// Causal depthwise conv1d (k=4) on xin half of xz, + bias, + SiLU.
// ---------------------------------------------------------------------------
__global__ __launch_bounds__(256)
void conv_silu_kernel(const float* __restrict__ xz, const float* __restrict__ cw,
                      const float* __restrict__ cb, float* __restrict__ out)
{
    const int idx = blockIdx.x * 256 + threadIdx.x;
    if (idx >= M_ROWS * D_INNERX) return;
    const int e = idx % D_INNERX;
    const int l = (idx / D_INNERX) % SEQX;
    const int b = idx / (D_INNERX * SEQX);

    const float* base = xz + (size_t)b * SEQX * (2 * D_INNERX) + e;
    float acc = cb[e];
#pragma unroll
    for (int t = 0; t < D_CONVX; ++t) {
        const int ls = l - (D_CONVX - 1) + t;
        if (ls >= 0) acc += cw[e * D_CONVX + t] * base[(size_t)ls * (2 * D_INNERX)];
    }
    out[idx] = acc / (1.0f + __expf(-acc));   // SiLU
}

// ---------------------------------------------------------------------------
// Selective scan. One thread per (b, channel e); 16-wide state in registers.
// B_t / C_t (shared by all channels) are staged through LDS in chunks of
// SCAN_CHUNK timesteps -> one barrier pair per 16 steps instead of per step.
// Next chunk is prefetched (global_prefetch_b8) while this one computes.
// Fuses the D skip-connection and the silu(z) gate.
// ---------------------------------------------------------------------------
__global__ __launch_bounds__(256)
void scan_kernel(const float* __restrict__ delta, const float* __restrict__ dbc,
                 const float* __restrict__ xin,   const float* __restrict__ xz,
                 const float* __restrict__ A_log, const float* __restrict__ Dp,
                 float* __restrict__ yact)
{
    __shared__ float sBC[SCAN_CHUNK * 32];    // per step: [0..15]=B_t, [16..31]=C_t
    const int b = blockIdx.y;
    const int e = blockIdx.x * 256 + threadIdx.x;

    float A[D_STATEX], h[D_STATEX];
#pragma unroll
    for (int n = 0; n < D_STATEX; ++n) {
        A[n] = -__expf(A_log[(size_t)e * D_STATEX + n]);
        h[n] = 0.0f;
    }
    const float  dpe   = Dp[e];
    const size_t rowD  = (size_t)b * SEQX * D_INNERX + e;
    const size_t rowZ  = (size_t)b * SEQX * (2 * D_INNERX) + D_INNERX + e;
    const size_t rowBC = (size_t)b * SEQX * DBC_W + DT_RANKX;

    // cooperative-load geometry: 256 threads x float2 = 512 floats = 16 steps
    const int li  = threadIdx.x * 2;
    const int srow = li >> 5;                 // 0..15  (timestep within chunk)
    const int scol = li & 31;                 // 0..30 even (B/C column)

    for (int t0 = 0; t0 < SEQX; t0 += SCAN_CHUNK) {
        __syncthreads();                      // previous chunk fully consumed
        {
            const v2f bc = *(const v2f*)(dbc + rowBC + (size_t)(t0 + srow) * DBC_W + scol);
            *(v2f*)&sBC[srow * 32 + scol] = bc;
        }
        __syncthreads();

        if (t0 + SCAN_CHUNK < SEQX) {         // warm L0/L2 for the next chunk
            __builtin_prefetch(dbc + rowBC + (size_t)(t0 + SCAN_CHUNK + srow) * DBC_W + scol, 0, 1);
            __builtin_prefetch(delta + rowD + (size_t)(t0 + SCAN_CHUNK) * D_INNERX, 0, 1);
            __builtin_prefetch(xin   + rowD + (size_t)(t0 + SCAN_CHUNK) * D_INNERX, 0, 1);
            __builtin_prefetch(xz    + rowZ + (size_t)(t0 + SCAN_CHUNK) * (2 * D_INNERX), 0, 1);
        }

        for (int tt = 0; tt < SCAN_CHUNK; ++tt) {
            const int   t  = t0 + tt;
            const float dt = delta[rowD + (size_t)t * D_INNERX];
            const float xv = xin  [rowD + (size_t)t * D_INNERX];
            const float zv = xz   [rowZ + (size_t)t * (2 * D_INNERX)];
            const float dx = dt * xv;
            const float* bcrow = &sBC[tt * 32];
            float y = 0.0f;
#pragma unroll
            for (int n = 0; n < D_STATEX; ++n) {
                const float dA = __expf(dt * A[n]);
                h[n] = dA * h[n] + dx * bcrow[n];
                y += h[n] * bcrow[16 + n];
            }
            y += xv * dpe;
            const float sz = zv / (1.0f + __expf(-zv));
            yact[rowD + (size_t)t * D_INNERX] = y * sz;
        }
    }
}

// ---------------------------------------------------------------------------
// Mean-pool over L.
// ---------------------------------------------------------------------------
__global__ __launch_bounds__(256)
void pool_kernel(const float* __restrict__ xn, float* __restrict__ pooled)
{
    const int idx = blockIdx.x * 256 + threadIdx.x;
    if (idx >= BATCHX * D_MODELX) return;
    const int b = idx / D_MODELX;
    const int d = idx % D_MODELX;
    const float* p = xn + (size_t)b * SEQX * D_MODELX + d;
    float s = 0.0f;
    for (int l = 0; l < SEQX; ++l) s += p[(size_t)l * D_MODELX];
    pooled[idx] = s * (1.0f / (float)SEQX);
}

// ---------------------------------------------------------------------------
// Tiny classifier head: logits (8x10), log-softmax, NLL loss.
// d_out[0..79] = logits, d_out[80] = loss.
// ---------------------------------------------------------------------------
__global__ __launch_bounds__(128)
void cls_kernel(const float* __restrict__ pooled, const float* __restrict__ cw,
                const float* __restrict__ cbias,  const int* __restrict__ labels,
                float* __restrict__ out)
{
    __shared__ float slog[BATCHX * NUM_CLS];
    const int t = threadIdx.x;
    if (t < BATCHX * NUM_CLS) {
        const int b = t / NUM_CLS;
        const int c = t % NUM_CLS;
        float s = cbias[c];
        const float* p = pooled + b * D_MODELX;
        const float* w = cw + c * D_MODELX;
        for (int d = 0; d < D_MODELX; ++d) s += p[d] * w[d];
        slog[t] = s;
        out[t]  = s;
    }
    __syncthreads();
    if (t == 0) {
        float loss = 0.0f;
        for (int b = 0; b < BATCHX; ++b) {
            float mx = slog[b * NUM_CLS];
            for (int c = 1; c < NUM_CLS; ++c) mx = fmaxf(mx, slog[b * NUM_CLS + c]);
            float se = 0.0f;
            for (int c = 0; c < NUM_CLS; ++c) se += __expf(slog[b * NUM_CLS + c] - mx);
            const float lse = mx + logf(se);
            const int   lab = labels[b];
            loss += lse - slog[b * NUM_CLS + lab];
        }
        out[BATCHX * NUM_CLS] = loss * (1.0f / (float)BATCHX);
    }
}

// ---------------------------------------------------------------------------
extern "C" void kernel_launch(void* const* d_in, const int* in_sizes, int n_in,
                              void* d_out, int out_size, void* d_ws, size_t ws_size,
                              hipStream_t stream)
{
    const float* x       = (const float*)d_in[0];
    const int*   labels  = (const int*)  d_in[1];
    const float* in_w    = (const float*)d_in[2];
    const float* conv_w  = (const float*)d_in[3];
    const float* conv_b  = (const float*)d_in[4];
    const float* xproj_w = (const float*)d_in[5];
    const float* dt_w    = (const float*)d_in[6];
    const float* dt_b    = (const float*)d_in[7];
    const float* A_log   = (const float*)d_in[8];
    const float* D_param = (const float*)d_in[9];
    const float* out_w   = (const float*)d_in[10];
    const float* norm_w  = (const float*)d_in[11];
    const float* normf_w = (const float*)d_in[12];
    const float* cls_w   = (const float*)d_in[13];
    const float* cls_b   = (const float*)d_in[14];

    char*  ws  = (char*)d_ws;
    size_t off = 0;
    auto alloc = [&](size_t bytes) -> float* {
        float* p = (float*)(ws + off);
        off += (bytes + 255) & ~(size_t)255;
        return p;
    };
    float* h      = alloc((size_t)M_ROWS * D_MODELX     * 4);
    float* xn     = alloc((size_t)M_ROWS * D_MODELX     * 4);
    float* xzbuf  = alloc((size_t)M_ROWS * 2 * D_INNERX * 4);
    float* xinc   = alloc((size_t)M_ROWS * D_INNERX     * 4);
    float* dbc    = alloc((size_t)M_ROWS * DBC_W        * 4);
    float* delta  = alloc((size_t)M_ROWS * D_INNERX     * 4);
    float* yact   = alloc((size_t)M_ROWS * D_INNERX     * 4);
    float* pooled = alloc((size_t)BATCHX * D_MODELX     * 4);

    hipMemcpyAsync(h, x, (size_t)M_ROWS * D_MODELX * 4,
                   hipMemcpyDeviceToDevice, stream);

    const dim3 gblk(32, 8);   // 8 waves; each wave owns a 16 x (16*NT) strip

    for (int i = 0; i < N_LAYERSX; ++i) {
        rmsnorm_kernel<<<M_ROWS, 256, 0, stream>>>(h, norm_w + i * D_MODELX, xn);

        // xz = xn @ in_proj_w^T : 16384x768 * 768x3072   (NT=4 -> N/64 tiles)
        gemm_wmma_f32<EPI_STORE, 4><<<dim3(M_ROWS / 128, (2 * D_INNERX) / 64), gblk, 0, stream>>>(
            xn, D_MODELX,
            in_w + (size_t)i * 2 * D_INNERX * D_MODELX, D_MODELX,
            xzbuf, 2 * D_INNERX, D_MODELX, nullptr);

        conv_silu_kernel<<<(M_ROWS * D_INNERX + 255) / 256, 256, 0, stream>>>(
            xzbuf, conv_w + (size_t)i * D_INNERX * D_CONVX,
            conv_b + i * D_INNERX, xinc);

        // dbc = xinc @ x_proj_w^T : 16384x1536 * 1536x80   (N=80 -> NT=1)
        gemm_wmma_f32<EPI_STORE, 1><<<dim3(M_ROWS / 128, DBC_W / 16), gblk, 0, stream>>>(
            xinc, D_INNERX,
            xproj_w + (size_t)i * DBC_W * D_INNERX, D_INNERX,
            dbc, DBC_W, D_INNERX, nullptr);

        // delta = softplus(dbc[:, :48] @ dt_w^T + dt_b) : 16384x48 * 48x1536
        gemm_wmma_f32<EPI_SOFTPLUS_BIAS, 4><<<dim3(M_ROWS / 128, D_INNERX / 64), gblk, 0, stream>>>(
            dbc, DBC_W,
            dt_w + (size_t)i * D_INNERX * DT_RANKX, DT_RANKX,
            delta, D_INNERX, DT_RANKX, dt_b + i * D_INNERX);

        scan_kernel<<<dim3(D_INNERX / 256, BATCHX), 256, 0, stream>>>(
            delta, dbc, xinc, xzbuf,
            A_log + (size_t)i * D_INNERX * D_STATEX,
            D_param + i * D_INNERX, yact);

        // h += yact @ out_proj_w^T : 16384x1536 * 1536x768 (residual fused)
        gemm_wmma_f32<EPI_RESID, 4><<<dim3(M_ROWS / 128, D_MODELX / 64), gblk, 0, stream>>>(
            yact, D_INNERX,
            out_w + (size_t)i * D_MODELX * D_INNERX, D_INNERX,
            h, D_MODELX, D_INNERX, nullptr);
    }

    rmsnorm_kernel<<<M_ROWS, 256, 0, stream>>>(h, normf_w, xn);
    pool_kernel<<<(BATCHX * D_MODELX + 255) / 256, 256, 0, stream>>>(xn, pooled);
    cls_kernel<<<1, 128, 0, stream>>>(pooled, cls_w, cls_b, labels, (float*)d_out);
}